// Model_11888469476064
// MI455X (gfx1250) — compile-verified
//
#include <hip/hip_runtime.h>
#include <hip/hip_bf16.h>

typedef __attribute__((ext_vector_type(2))) float v2f;
typedef __attribute__((ext_vector_type(8))) float v8f;

// Problem constants
#define NB   4
#define CIN  64
#define CMID 128
#define H    128
#define W    128
#define PIX  (H * W)          // 16384
#define PH   64               // pooled dims
#define PW   64
#define PPIX (PH * PW)        // 4096

// Workspace layout (in floats)
#define PROJ_OFF  0                         // 4*128*16384
#define ATT_OFF   (PROJ_OFF + NB*CMID*PIX)  // 4*128*4096
#define ATTD_OFF  (ATT_OFF + NB*CMID*PPIX)  // 4*128*16384
#define CS_OFF    (ATTD_OFF + NB*CMID*PIX)  // 4*16384
#define GATE_OFF  (CS_OFF + NB*PIX)
#define GMUL_OFF  (GATE_OFF + NB*PIX)
#define WS_FLOATS (GMUL_OFF + NB*PIX)

// ---------------------------------------------------------------------------
// K1: proj[n,o,p] = sum_c Wp[o,c] * x[n,c,p] + bp[o]
// One wave owns a 16-pixel column and ALL 128 output channels:
// 8 accumulators, B (x pixels) loaded once per k-step and reused 8x.
// A layout (16x4 f32): lanes 0-15: M=lane, regs {K0,K1}; lanes 16-31: M=lane-16, regs {K2,K3}
// B layout (4x16 f32): lanes 0-15: N=lane, regs {K0,K1}; lanes 16-31: N=lane-16, regs {K2,K3}
// D layout: reg r, lanes 0-15 -> M=r, N=lane; lanes 16-31 -> M=r+8, N=lane-16
// ---------------------------------------------------------------------------
__global__ __launch_bounds__(256) void proj_gemm_kernel(
    const float* __restrict__ x, const float* __restrict__ Wp,
    const float* __restrict__ bp, float* __restrict__ proj)
{
    const int lane = threadIdx.x & 31;
    const int tile = blockIdx.x * 8 + (threadIdx.x >> 5);   // 4096 tiles: (n, ptile)
    const int pt = tile & 1023;           // pixel tile (16 px)
    const int n  = tile >> 10;
    const int p0 = pt << 4;
    const int m  = lane & 15;
    const int kb = (lane >> 4) << 1;      // 0 or 2

    const float* xb = x + (size_t)n * CIN * PIX + p0 + m;
    v8f acc[8] = {};
#pragma unroll
    for (int c0 = 0; c0 < CIN; c0 += 4) {
        v2f b;
        b.x = xb[(size_t)(c0 + kb + 0) * PIX];
        b.y = xb[(size_t)(c0 + kb + 1) * PIX];
#pragma unroll
        for (int ot = 0; ot < 8; ++ot) {
            v2f a;
            const int row = (ot << 4) + m;
            a.x = Wp[row * CIN + c0 + kb + 0];
            a.y = Wp[row * CIN + c0 + kb + 1];
            acc[ot] = __builtin_amdgcn_wmma_f32_16x16x4_f32(
                false, a, false, b, (short)0, acc[ot], false, false);
        }
    }
    float* pb = proj + (size_t)n * CMID * PIX + p0 + (lane & 15);
    const int rbase = (lane >> 4) << 3;
#pragma unroll
    for (int ot = 0; ot < 8; ++ot) {
#pragma unroll
        for (int r = 0; r < 8; ++r) {
            const int row = (ot << 4) + rbase + r;
            pb[(size_t)row * PIX] = acc[ot][r] + bp[row];
        }
    }
}

// ---------------------------------------------------------------------------
// K2: LPPool2d(p=2, 2x2/2) then softmax over 128 channels.
// Block = 64 threads = one pooled row (y2); thread = x2. Per-thread LDS column.
// ---------------------------------------------------------------------------
__global__ __launch_bounds__(64) void pool_softmax_kernel(
    const float* __restrict__ proj, float* __restrict__ att)
{
    __shared__ float sv[CMID * 64];
    const int t  = threadIdx.x;          // x2
    const int y2 = blockIdx.x & 63;
    const int n  = blockIdx.x >> 6;

    const float* pb = proj + (size_t)n * CMID * PIX + (size_t)(2 * y2) * W + 2 * t;
    float mx = -3.4e38f;
    for (int c = 0; c < CMID; ++c) {
        const float* p = pb + (size_t)c * PIX;
        const float a = p[0], b = p[1], cc = p[W], d = p[W + 1];
        const float v = sqrtf(a * a + b * b + cc * cc + d * d);
        sv[c * 64 + t] = v;
        mx = fmaxf(mx, v);
    }
    float s = 0.0f;
    for (int c = 0; c < CMID; ++c) {
        const float e = __expf(sv[c * 64 + t] - mx);
        sv[c * 64 + t] = e;
        s += e;
    }
    const float inv = 1.0f / s;
    float* ab = att + (size_t)n * CMID * PPIX + y2 * PW + t;
    for (int c = 0; c < CMID; ++c)
        ab[(size_t)c * PPIX] = sv[c * 64 + t] * inv;
}

// ---------------------------------------------------------------------------
// K3: att_up (bilinear x2, align_corners=False), attended = proj*att_up,
//     chansum = sum_c attended.  One thread per output pixel.
// ---------------------------------------------------------------------------
__global__ __launch_bounds__(256) void attend_kernel(
    const float* __restrict__ proj, const float* __restrict__ att,
    float* __restrict__ attended, float* __restrict__ chansum)
{
    const int gid = blockIdx.x * 256 + threadIdx.x;     // 65536 px
    const int w = gid & 127;
    const int h = (gid >> 7) & 127;
    const int n = gid >> 14;

    const float ry = fmaxf((float)h * 0.5f - 0.25f, 0.0f);
    const float rx = fmaxf((float)w * 0.5f - 0.25f, 0.0f);
    const int y0 = (int)ry, x0 = (int)rx;
    const int y1 = min(y0 + 1, PH - 1), x1 = min(x0 + 1, PW - 1);
    const float fy = ry - (float)y0, fx = rx - (float)x0;
    const float w00 = (1.0f - fy) * (1.0f - fx), w01 = (1.0f - fy) * fx;
    const float w10 = fy * (1.0f - fx),          w11 = fy * fx;
    const int i00 = y0 * PW + x0, i01 = y0 * PW + x1;
    const int i10 = y1 * PW + x0, i11 = y1 * PW + x1;

    const float* ab = att + (size_t)n * CMID * PPIX;
    const float* pb = proj + (size_t)n * CMID * PIX + (size_t)h * W + w;
    float* ob = attended + (size_t)n * CMID * PIX + (size_t)h * W + w;

    float csum = 0.0f;
    for (int c = 0; c < CMID; ++c) {
        const float* a = ab + (size_t)c * PPIX;
        const float au = w00 * a[i00] + w01 * a[i01] + w10 * a[i10] + w11 * a[i11];
        const float v = pb[(size_t)c * PIX] * au;
        ob[(size_t)c * PIX] = v;
        csum += v;
    }
    chansum[gid] = csum;
}

// ---------------------------------------------------------------------------
// K4a: gate = sigmoid( zero-padded 3x3 sum of chansum / (128*9) )
// ---------------------------------------------------------------------------
__global__ __launch_bounds__(256) void gate_kernel(
    const float* __restrict__ chansum, float* __restrict__ gate)
{
    const int gid = blockIdx.x * 256 + threadIdx.x;
    const int w = gid & 127;
    const int h = (gid >> 7) & 127;
    const int n = gid >> 14;
    const float* cb = chansum + (size_t)n * PIX;
    float s = 0.0f;
#pragma unroll
    for (int dy = -1; dy <= 1; ++dy) {
        const int y = h + dy;
        if ((unsigned)y < (unsigned)H) {
#pragma unroll
            for (int dx = -1; dx <= 1; ++dx) {
                const int xx = w + dx;
                if ((unsigned)xx < (unsigned)W) s += cb[y * W + xx];
            }
        }
    }
    const float m = s * (1.0f / (float)(CMID * 9));
    gate[gid] = 1.0f / (1.0f + __expf(-m));
}

// ---------------------------------------------------------------------------
// K4b: G[y,x] = clipped 3x3 sum of gate (fold overlap multiplicity)
// ---------------------------------------------------------------------------
__global__ __launch_bounds__(256) void gmul_kernel(
    const float* __restrict__ gate, float* __restrict__ G)
{
    const int gid = blockIdx.x * 256 + threadIdx.x;
    const int w = gid & 127;
    const int h = (gid >> 7) & 127;
    const int n = gid >> 14;
    const float* gb = gate + (size_t)n * PIX;
    float s = 0.0f;
#pragma unroll
    for (int dy = -1; dy <= 1; ++dy) {
        const int y = h + dy;
        if ((unsigned)y < (unsigned)H) {
#pragma unroll
            for (int dx = -1; dx <= 1; ++dx) {
                const int xx = w + dx;
                if ((unsigned)xx < (unsigned)W) s += gb[y * W + xx];
            }
        }
    }
    G[gid] = s;
}

// ---------------------------------------------------------------------------
// K5: out[n,co,p] = sum_o Wr[co,o] * (attended[n,o,p] * G[n,p]) + br[co] + x[n,co,p]
// One wave owns a 16-pixel column and ALL 64 output channels:
// 4 accumulators; B (attended*G) loaded once per k-step, reused 4x.
// ---------------------------------------------------------------------------
__global__ __launch_bounds__(256) void out_gemm_kernel(
    const float* __restrict__ attended, const float* __restrict__ G,
    const float* __restrict__ Wr, const float* __restrict__ br,
    const float* __restrict__ x, float* __restrict__ out)
{
    const int lane = threadIdx.x & 31;
    const int tile = blockIdx.x * 8 + (threadIdx.x >> 5);   // 4096 tiles: (n, ptile)
    const int pt = tile & 1023;
    const int n  = tile >> 10;
    const int p0 = pt << 4;
    const int m  = lane & 15;
    const int kb = (lane >> 4) << 1;

    const float* atb = attended + (size_t)n * CMID * PIX + p0 + m;
    const float g = G[n * PIX + p0 + m];

    v8f acc[4] = {};
#pragma unroll
    for (int c0 = 0; c0 < CMID; c0 += 4) {
        v2f b;
        b.x = atb[(size_t)(c0 + kb + 0) * PIX] * g;
        b.y = atb[(size_t)(c0 + kb + 1) * PIX] * g;
#pragma unroll
        for (int ot = 0; ot < 4; ++ot) {
            v2f a;
            const int row = (ot << 4) + m;
            a.x = Wr[row * CMID + c0 + kb + 0];
            a.y = Wr[row * CMID + c0 + kb + 1];
            acc[ot] = __builtin_amdgcn_wmma_f32_16x16x4_f32(
                false, a, false, b, (short)0, acc[ot], false, false);
        }
    }
    const float* xb = x + (size_t)n * CIN * PIX + p0 + (lane & 15);
    float* ob = out + (size_t)n * CIN * PIX + p0 + (lane & 15);
    const int rbase = (lane >> 4) << 3;
#pragma unroll
    for (int ot = 0; ot < 4; ++ot) {
#pragma unroll
        for (int r = 0; r < 8; ++r) {
            const int row = (ot << 4) + rbase + r;
            ob[(size_t)row * PIX] = acc[ot][r] + br[row] + xb[(size_t)row * PIX];
        }
    }
}

// ---------------------------------------------------------------------------
extern "C" void kernel_launch(void* const* d_in, const int* in_sizes, int n_in,
                              void* d_out, int out_size, void* d_ws, size_t ws_size,
                              hipStream_t stream)
{
    const float* x  = (const float*)d_in[0];
    const float* Wp = (const float*)d_in[1];
    const float* bp = (const float*)d_in[2];
    const float* Wr = (const float*)d_in[3];
    const float* br = (const float*)d_in[4];
    float* out = (float*)d_out;
    float* ws  = (float*)d_ws;

    if (ws_size < (size_t)WS_FLOATS * sizeof(float)) return;

    float* proj     = ws + PROJ_OFF;
    float* att      = ws + ATT_OFF;
    float* attended = ws + ATTD_OFF;
    float* chansum  = ws + CS_OFF;
    float* gate     = ws + GATE_OFF;
    float* G        = ws + GMUL_OFF;

    // K1: 4096 wave-tiles (n x 1024 pixel tiles), 8 waves per block
    proj_gemm_kernel<<<512, 256, 0, stream>>>(x, Wp, bp, proj);
    // K2: one block (64 thr) per (n, y2) row: 4*64 = 256 blocks
    pool_softmax_kernel<<<256, 64, 0, stream>>>(proj, att);
    // K3: 65536 pixels
    attend_kernel<<<256, 256, 0, stream>>>(proj, att, attended, chansum);
    // K4: stencils
    gate_kernel<<<256, 256, 0, stream>>>(chansum, gate);
    gmul_kernel<<<256, 256, 0, stream>>>(gate, G);
    // K5: 4096 wave-tiles, 8 waves per block
    out_gemm_kernel<<<512, 256, 0, stream>>>(attended, G, Wr, br, x, out);
}